// ProxyNCALoss_7258494730645
// MI455X (gfx1250) — compile-verified
//
#include <hip/hip_runtime.h>
#include <math.h>

typedef float v2f __attribute__((ext_vector_type(2)));
typedef float v8f __attribute__((ext_vector_type(8)));

#define BS       1024
#define DIM      128
#define P_TOTAL  100000
#define CHUNK    2048
#define NC       ((P_TOTAL + CHUNK - 1) / CHUNK)   /* 49 */
#define LDS_STRIDE 130      /* 128 + 2 pad: conflict-free b64 column reads */
#define RED_STRIDE 20       /* 16 + 4 pad: conflict-free b128 row reads, 16B aligned */
#define LOG2E    1.4426950408889634f

__device__ __forceinline__ float dot4(const float4& a, const float4& b) {
    return a.x*b.x + a.y*b.y + a.z*b.z + a.w*b.w;
}

// 1 / max(sqrt(n2), 1e-12)  ==  rsqrt(max(n2, 1e-24)); single v_rsq_f32.
__device__ __forceinline__ float inv_norm(float n2) {
    return __builtin_amdgcn_rsqf(fmaxf(n2, 1e-24f));
}

// ---------------------------------------------------------------------------
// Kernel 1: fused GEMM (f32 WMMA 16x16x4) + partial sum(exp(-d)) per row.
// grid = (NC, BS/128), block = 256 (8 waves; wave w owns rows blk*128+w*16..+15).
// Proxy normalization is deferred past the GEMM: WMMA accumulates raw proxy
// data; per-column scale folds into the exp2 argument (1 FMA per element).
// ---------------------------------------------------------------------------
__global__ __launch_bounds__(256) void pnca_gemm_lse(
    const float* __restrict__ batch,
    const float* __restrict__ proxies,
    const int*   __restrict__ labels,
    float*       __restrict__ partial_s)   // [NC][BS]
{
    __shared__ __align__(16) float lds_p[16 * LDS_STRIDE];  // raw proxy tile
    __shared__ __align__(16) float lds_red[16 * RED_STRIDE];// sumsq partials

    const int tid  = threadIdx.x;
    const int wave = tid >> 5;
    const int lane = tid & 31;
    const int ln16 = lane & 15;
    const int hi   = lane >> 4;              // 0: lanes 0-15, 1: lanes 16-31

    // ---- normalize this wave's 16 batch rows into WMMA A fragments ----
    const int rb  = blockIdx.y * 128 + wave * 16;
    const int row = rb + ln16;
    const float* brow = batch + (size_t)row * DIM;

    v2f a[32];
    float n2 = 0.f;
#pragma unroll
    for (int kb = 0; kb < 32; ++kb) {
        // A 16x4 layout: VGPR0 = K {0 low / 2 high}, VGPR1 = K {1 low / 3 high}
        v2f t = *(const v2f*)(brow + 4 * kb + 2 * hi);
        a[kb] = t;
        n2 += t.x * t.x + t.y * t.y;         // each lane covers half the row
    }
    n2 += __shfl_xor(n2, 16);                // lanes L and L+16 complete the row
    {
        float inv = inv_norm(n2);
        float scl = 3.0f * inv;
        float b2  = n2 * scl * scl;
#pragma unroll
        for (int kb = 0; kb < 32; ++kb) { a[kb].x *= scl; a[kb].y *= scl; }
        n2 = b2;                             // reuse register as b2
    }
    const float b2_lane  = n2;
    const int   lab_lane = labels[row];

    // C/D tile layout: VGPR v -> row (v + 8*hi), column = ln16.
    float offb[8]; int labv[8];
#pragma unroll
    for (int v = 0; v < 8; ++v) {
        offb[v] = -LOG2E * __shfl(b2_lane, v + 8 * hi);  // -b2(row)*log2(e)
        labv[v] = __shfl(lab_lane, v + 8 * hi);
    }

    float sacc[8] = {0.f, 0.f, 0.f, 0.f, 0.f, 0.f, 0.f, 0.f};

    const int jstart = blockIdx.x * CHUNK;
    const int jend   = min(jstart + CHUNK, P_TOTAL);   // P_TOTAL % 16 == 0

    // ---- software-pipelined proxy staging ----
    const int pxy = tid >> 4;                // proxy within tile
    const int seg = tid & 15;                // 8-element segment
    const float* pbase = proxies + (size_t)pxy * DIM + seg * 8;

    float4 q0 = *(const float4*)(pbase + (size_t)jstart * DIM);
    float4 q1 = *(const float4*)(pbase + (size_t)jstart * DIM + 4);

    for (int j0 = jstart; j0 < jend; j0 += 16) {
        // commit current tile to LDS
        lds_red[pxy * RED_STRIDE + seg] = dot4(q0, q0) + dot4(q1, q1);
        float* dst = &lds_p[pxy * LDS_STRIDE + seg * 8];
        *(float4*)(dst)     = q0;
        *(float4*)(dst + 4) = q1;
        __syncthreads();

        // prefetch next tile (clamped in-bounds on the final iteration)
        const int jn = (j0 + 16 < jend) ? (j0 + 16) : jstart;
        q0 = *(const float4*)(pbase + (size_t)jn * DIM);
        q1 = *(const float4*)(pbase + (size_t)jn * DIM + 4);

        // per-lane norm of this lane's column (4x b128 reads, padded stride)
        float n2c = 0.f;
#pragma unroll
        for (int i = 0; i < 4; ++i) {
            float4 t = *(const float4*)&lds_red[ln16 * RED_STRIDE + 4 * i];
            n2c += t.x + t.y + t.z + t.w;
        }
        const float inv = inv_norm(n2c);             // v_rsq_f32
        const float psc = 3.0f * inv;                // proxy scale
        const float p2c = n2c * psc * psc;           // |3*normalize(p)|^2
        const float tw2 = (2.0f * LOG2E) * psc;      // 2*psc*log2(e)
        const float o2  = -LOG2E * p2c;              // -p2*log2(e)

        // ---- 16x16x128 tile: pure ds_load + 32x V_WMMA_F32_16X16X4_F32 ----
        v8f c = {};
        const float* bsrc = &lds_p[ln16 * LDS_STRIDE + 2 * hi];
#pragma unroll
        for (int kb = 0; kb < 32; ++kb) {
            v2f bf = *(const v2f*)(bsrc + 4 * kb);
            c = __builtin_amdgcn_wmma_f32_16x16x4_f32(
                    false, a[kb], false, bf, (short)0, c, false, false);
        }

        // ---- epilogue: exp2(2*psc*S*log2e - (b2+p2)*log2e), label masked ----
        const int col = j0 + ln16;
#pragma unroll
        for (int v = 0; v < 8; ++v) {
            float arg = __builtin_fmaf(c[v], tw2, offb[v] + o2);
            float e   = __builtin_amdgcn_exp2f(arg);
            sacc[v] += (labv[v] == col) ? 0.0f : e;
        }
        __syncthreads();   // protect LDS tile before next iteration's stores
    }

    // ---- reduce the 16 column-lanes of each half-wave ----
#pragma unroll
    for (int v = 0; v < 8; ++v) {
        float s = sacc[v];
        s += __shfl_xor(s, 1);
        s += __shfl_xor(s, 2);
        s += __shfl_xor(s, 4);
        s += __shfl_xor(s, 8);
        sacc[v] = s;
    }
    if (ln16 == 0) {
#pragma unroll
        for (int v = 0; v < 8; ++v)
            partial_s[(size_t)blockIdx.x * BS + rb + 8 * hi + v] = sacc[v];
    }
}

// ---------------------------------------------------------------------------
// Kernel 2: d_pos[r]; one wave per row; grid = BS/8 blocks of 256.
// ---------------------------------------------------------------------------
__global__ __launch_bounds__(256) void pnca_dpos(
    const float* __restrict__ batch,
    const float* __restrict__ proxies,
    const int*   __restrict__ labels,
    float*       __restrict__ d_pos)
{
    const int wave = threadIdx.x >> 5;
    const int lane = threadIdx.x & 31;
    const int r    = blockIdx.x * 8 + wave;
    const int lab  = labels[r];

    float4 bx = *(const float4*)(batch   + (size_t)r   * DIM + lane * 4);
    float4 px = *(const float4*)(proxies + (size_t)lab * DIM + lane * 4);

    float n2b = dot4(bx, bx);
    float n2p = dot4(px, px);
    float dp  = dot4(bx, px);
#pragma unroll
    for (int m = 1; m < 32; m <<= 1) {
        n2b += __shfl_xor(n2b, m);
        n2p += __shfl_xor(n2p, m);
        dp  += __shfl_xor(dp,  m);
    }
    if (lane == 0) {
        float ib = inv_norm(n2b);            // 1/max(|b|,eps), one v_rsq
        float ip = inv_norm(n2p);
        d_pos[r] = 9.0f * n2b * ib * ib
                 + 9.0f * n2p * ip * ip
                 - 18.0f * dp * ib * ip;
    }
}

// ---------------------------------------------------------------------------
// Kernel 3: combine partials, lse = log(sum), mean over rows. 1 block, 1024 thr.
// ---------------------------------------------------------------------------
__global__ __launch_bounds__(1024) void pnca_final(
    const float* __restrict__ partial_s,
    const float* __restrict__ d_pos,
    float*       __restrict__ out)
{
    __shared__ float red[1024];
    const int r = threadIdx.x;
    float s = 0.f;
    for (int c = 0; c < NC; ++c) s += partial_s[(size_t)c * BS + r];
    red[r] = d_pos[r] + logf(s);
    __syncthreads();
    for (int off = 512; off > 0; off >>= 1) {
        if (r < off) red[r] += red[r + off];
        __syncthreads();
    }
    if (r == 0) out[0] = red[0] * (1.0f / BS);
}

// ---------------------------------------------------------------------------
extern "C" void kernel_launch(void* const* d_in, const int* in_sizes, int n_in,
                              void* d_out, int out_size, void* d_ws, size_t ws_size,
                              hipStream_t stream)
{
    (void)in_sizes; (void)n_in; (void)out_size; (void)ws_size;
    const float* batch   = (const float*)d_in[0];
    const float* proxies = (const float*)d_in[1];
    const int*   labels  = (const int*)d_in[2];
    float*       out     = (float*)d_out;

    float* d_pos    = (float*)d_ws;                       // 1024 floats
    float* partials = (float*)((char*)d_ws + 4096);       // NC*1024 floats (~200 KB)

    dim3 g1(NC, BS / 128);
    pnca_gemm_lse<<<g1, 256, 0, stream>>>(batch, proxies, labels, partials);
    pnca_dpos<<<BS / 8, 256, 0, stream>>>(batch, proxies, labels, d_pos);
    pnca_final<<<1, 1024, 0, stream>>>(partials, d_pos, out);
}